// myConv_input_49220325212422
// MI455X (gfx1250) — compile-verified
//
#include <hip/hip_runtime.h>

typedef __attribute__((ext_vector_type(2))) float v2f;
typedef __attribute__((ext_vector_type(8))) float v8f;

#define KSZ   3
#define CIN   3
#define COUT  128
#define IH    256
#define IW    256
#define OH    254
#define OW    254
#define NBATCH 8
#define KTOT  27                 // 3*3*3 contraction length
#define KPAD  28                 // padded to multiple of 4 for 16x16x4 WMMA
#define BSTRIDE 132              // LDS row stride (128 + 4 pad: kills bank conflicts)
#define TOTAL_PIX (NBATCH * OH * OW)      // 516128
#define MU_ELEMS  ((size_t)TOTAL_PIX * COUT)

__global__ __launch_bounds__(256)
void bayes_conv_wmma_f32(const float* __restrict__ in,
                         const float* __restrict__ w_mu,
                         const float* __restrict__ w_sigma,
                         float* __restrict__ out)
{
    __shared__ float ldsB[KPAD * BSTRIDE];

    const int tid = threadIdx.x;

    // Stage weights (27x128 f32, HWIO flat = k*128+co) into LDS, zero-pad row 27.
    for (int i = tid; i < KPAD * COUT; i += 256) {
        int k  = i >> 7;
        int co = i & 127;
        ldsB[k * BSTRIDE + co] = (k < KTOT) ? w_mu[i] : 0.0f;
    }
    __syncthreads();

    const int lane = tid & 31;          // wave32
    const int wave = tid >> 5;          // 8 waves / block
    const int half = lane >> 4;         // 0: lanes 0-15, 1: lanes 16-31
    const int nidx = lane & 15;         // N column (and A-matrix M row) index

    const int strip = blockIdx.x * 128 + wave * 16;   // first pixel of this wave's M strip

    // Pixel this lane gathers A-rows for (clamped so EXEC stays all-1s for WMMA).
    int pix  = strip + nidx;
    int pixc = (pix < TOTAL_PIX) ? pix : (TOTAL_PIX - 1);
    int ow = pixc % OW;
    int t  = pixc / OW;
    int oh = t % OH;
    int n  = t / OH;
    const int ibase = ((n * IH + oh) * IW + ow) * CIN;

    v8f acc[8];
#pragma unroll
    for (int nt = 0; nt < 8; ++nt) acc[nt] = (v8f){0,0,0,0,0,0,0,0};

    float ssq = 0.0f;   // per-lane partial sum of x^2 over this lane's K subset

    // 7 WMMA K-steps cover K = 0..27 (27 real + 1 zero pad).
#pragma unroll
    for (int s = 0; s < 7; ++s) {
        const int k0 = 4 * s + 2 * half;   // A layout: VGPR0 = K=k0, VGPR1 = K=k0+1
        const int k1 = k0 + 1;

        // im2col: k -> (kh, kw, c), c fastest (matches HWIO weight flattening)
        int kh0 = k0 / 9, r0 = k0 - 9 * kh0, kw0 = r0 / 3, c0 = r0 - 3 * kw0;
        int kh1 = k1 / 9, r1 = k1 - 9 * kh1, kw1 = r1 / 3, c1 = r1 - 3 * kw1;

        float a0 = in[ibase + kh0 * (IW * CIN) + kw0 * CIN + c0];
        float a1 = (k1 < KTOT) ? in[ibase + kh1 * (IW * CIN) + kw1 * CIN + c1] : 0.0f;
        ssq += a0 * a0 + a1 * a1;

        v2f afrag;
        afrag.x = a0;
        afrag.y = a1;

#pragma unroll
        for (int nt = 0; nt < 8; ++nt) {
            // B 4x16 fragment: lane holds col nidx, rows k0 / k0+1 (mirrors A halves)
            v2f bfrag;
            bfrag.x = ldsB[k0 * BSTRIDE + nt * 16 + nidx];
            bfrag.y = ldsB[k1 * BSTRIDE + nt * 16 + nidx];
            acc[nt] = __builtin_amdgcn_wmma_f32_16x16x4_f32(
                false, afrag, false, bfrag, (short)0, acc[nt], false, false);
        }
    }

    // ---- mu output: D layout -> lane holds col nidx, rows v + 8*half ----
#pragma unroll
    for (int v = 0; v < 8; ++v) {
        int p = strip + v + 8 * half;
        if (p < TOTAL_PIX) {
            float* o = out + (size_t)p * COUT + nidx;
#pragma unroll
            for (int nt = 0; nt < 8; ++nt)
                o[nt * 16] = acc[nt][v];
        }
    }

    // ---- sigma output: ssq(pixel) * softplus(w_sigma)[cout] ----
    // Lanes L and L+16 together cover all K for pixel L%16 -> combine halves.
    ssq += __shfl_xor(ssq, 16, 32);

    // softplus for 4 consecutive couts per lane (32 lanes * 4 = 128)
    const float* ws = w_sigma + lane * 4;
    float4 sp;
    sp.x = log1pf(__expf(ws[0]));
    sp.y = log1pf(__expf(ws[1]));
    sp.z = log1pf(__expf(ws[2]));
    sp.w = log1pf(__expf(ws[3]));

    float* out_sigma = out + MU_ELEMS;
#pragma unroll
    for (int p16 = 0; p16 < 16; ++p16) {
        float sq = __shfl(ssq, p16, 32);   // broadcast pixel p16's sum-of-squares
        int p = strip + p16;
        if (p < TOTAL_PIX) {
            float4 v4;
            v4.x = sq * sp.x;
            v4.y = sq * sp.y;
            v4.z = sq * sp.z;
            v4.w = sq * sp.w;
            *reinterpret_cast<float4*>(out_sigma + (size_t)p * COUT + lane * 4) = v4;
        }
    }
}

extern "C" void kernel_launch(void* const* d_in, const int* in_sizes, int n_in,
                              void* d_out, int out_size, void* d_ws, size_t ws_size,
                              hipStream_t stream)
{
    const float* in      = (const float*)d_in[0];
    const float* w_mu    = (const float*)d_in[1];
    const float* w_sigma = (const float*)d_in[2];
    float* out           = (float*)d_out;

    // 128 pixels (8 waves x 16) per block
    dim3 grid((TOTAL_PIX + 127) / 128);
    bayes_conv_wmma_f32<<<grid, 256, 0, stream>>>(in, w_mu, w_sigma, out);
}